// ModelNew_70995809403125
// MI455X (gfx1250) — compile-verified
//
#include <hip/hip_runtime.h>
#include <math.h>

typedef __attribute__((ext_vector_type(2))) float v2f;
typedef __attribute__((ext_vector_type(8))) float v8f;

#define T_STEPS 4096
#define BATCH   64
#define IDIM    128
#define HDIM    256
#define NWG     16        // one WG per 16 h-columns; each on its own WGP
#define NTHREADS 384      // 12 wave32 = 3 gates x 4 row-tiles
#define LPAD    2         // LDS row padding (floats) for weight slices
#define HPAD    4         // LDS row padding (floats) for h stage (16B-aligned rows)

// ---------------------------------------------------------------------------
// init: zero the grid-barrier counter, copy h0 into the ping buffer
// ---------------------------------------------------------------------------
__global__ void gru_init_kernel(const float* __restrict__ h0,
                                float* __restrict__ hbuf,
                                unsigned* __restrict__ ctr) {
    int i = blockIdx.x * blockDim.x + threadIdx.x;
    if (i == 0) *ctr = 0u;
    if (i < BATCH * HDIM) hbuf[i] = h0[i];
}

// ---------------------------------------------------------------------------
// scan: 16 WGs cooperatively run the GRU recurrence with fp32 WMMA.
// Per step: async-stage h into LDS (overlapped with the x-projection GEMM),
// recurrent GEMM entirely from LDS, gate fusion, device-wide barrier.
// ---------------------------------------------------------------------------
__launch_bounds__(NTHREADS)
__global__ void gru_scan_kernel(const float* __restrict__ x,     // [T,B,I]
                                const float* __restrict__ w_ih,  // [3H,I]
                                const float* __restrict__ w_hh,  // [3H,H]
                                const float* __restrict__ b_ih,  // [3H]
                                const float* __restrict__ b_hh,  // [3H]
                                float* __restrict__ h0buf,       // [B,H] ping
                                float* __restrict__ h1buf,       // [B,H] pong
                                unsigned* __restrict__ ctr,
                                float* __restrict__ out) {       // [T,B,H]
    extern __shared__ float smem[];
    float* lds_wih = smem;                                   // 48 x (IDIM+LPAD)
    float* lds_whh = lds_wih + 48 * (IDIM + LPAD);           // 48 x (HDIM+LPAD)
    float* lds_h   = lds_whh + 48 * (HDIM + LPAD);           // 64 x (HDIM+HPAD)
    float* lds_gi  = lds_h + BATCH * (HDIM + HPAD);          // 3 x 64 x 16
    float* lds_gh  = lds_gi + 3 * BATCH * 16;                // 3 x 64 x 16

    const int tid   = threadIdx.x;
    const int g     = blockIdx.x;       // owns h columns [g*16, g*16+16)
    const int wave  = tid >> 5;
    const int lane  = tid & 31;
    const int gate  = wave >> 2;        // 0=r, 1=z, 2=n
    const int mtile = wave & 3;         // batch rows m0..m0+15
    const int m0    = mtile * 16;
    const int nlo   = lane & 15;        // M (for A) / N (for B,C,D) sub-index
    const int hi    = lane >> 4;
    const int koff  = 2 * hi;           // K offset of this half-wave's fragment

    // ---- stage this WG's weight slices into LDS (reused 4096x) ----
    for (int idx = tid; idx < 48 * IDIM; idx += NTHREADS) {
        int row = idx / IDIM, col = idx - row * IDIM;
        int gg = row >> 4, nn = row & 15;
        lds_wih[row * (IDIM + LPAD) + col] =
            w_ih[(size_t)(gg * HDIM + g * 16 + nn) * IDIM + col];
    }
    for (int idx = tid; idx < 48 * HDIM; idx += NTHREADS) {
        int row = idx / HDIM, col = idx - row * HDIM;
        int gg = row >> 4, nn = row & 15;
        lds_whh[row * (HDIM + LPAD) + col] =
            w_hh[(size_t)(gg * HDIM + g * 16 + nn) * HDIM + col];
    }
    const float bi = b_ih[gate * HDIM + g * 16 + nlo];
    const float bh = b_hh[gate * HDIM + g * 16 + nlo];
    __syncthreads();

    const float* wih_row = &lds_wih[(gate * 16 + nlo) * (IDIM + LPAD)];
    const float* whh_row = &lds_whh[(gate * 16 + nlo) * (HDIM + LPAD)];
    const float* hrow_l  = &lds_h[(m0 + nlo) * (HDIM + HPAD)];

    for (int t = 0; t < T_STEPS; ++t) {
        const float* hcur  = (t & 1) ? h1buf : h0buf;
        float*       hnext = (t & 1) ? h0buf : h1buf;

        // ---- kick async DMA of h_t (64KB) into LDS; overlaps the gi GEMM ----
        // 4096 chunks of 16B; rows padded to 16B multiples (HPAD=4 floats).
        for (int c = tid; c < BATCH * (HDIM / 4); c += NTHREADS) {
            int row = c >> 6;                 // 64 chunks of b128 per row
            int part = c & 63;
            const float* gsrc = hcur + (size_t)row * HDIM + part * 4;
            unsigned ldst = (unsigned)(uintptr_t)(lds_h + row * (HDIM + HPAD) + part * 4);
            asm volatile("global_load_async_to_lds_b128 %0, %1, off"
                         :: "v"(ldst), "v"(gsrc) : "memory");
        }

        // ---- input projection slice: acc_i = x_t[m0:m0+16,:] . Wih_sliceT
        v8f acc_i = {};
        const float* xrow = x + ((size_t)t * BATCH + m0 + nlo) * IDIM;
        if (t + 1 < T_STEPS)
            __builtin_prefetch(x + ((size_t)(t + 1) * BATCH + m0 + nlo) * IDIM, 0, 1);
        #pragma unroll 8
        for (int k = 0; k < IDIM; k += 4) {
            v2f a = *(const v2f*)(xrow + k + koff);      // A[m=nlo][koff..koff+1]
            v2f b = *(const v2f*)(wih_row + k + koff);   // B[koff..][n=nlo]
            acc_i = __builtin_amdgcn_wmma_f32_16x16x4_f32(
                false, a, false, b, (short)0, acc_i, false, false);
        }

        // ---- drain the async h copy, then everyone can read lds_h ----
        asm volatile("s_wait_asynccnt 0x0" ::: "memory");
        __syncthreads();

        // ---- recurrent slice from LDS: acc_h = h[m0:m0+16,:] . Whh_sliceT
        v8f acc_h = {};
        #pragma unroll 8
        for (int k = 0; k < HDIM; k += 4) {
            v2f a = *(const v2f*)(hrow_l + k + koff);
            v2f b = *(const v2f*)(whh_row + k + koff);
            acc_h = __builtin_amdgcn_wmma_f32_16x16x4_f32(
                false, a, false, b, (short)0, acc_h, false, false);
        }

        // ---- stash per-gate partials in LDS (C/D layout: row = r + 8*hi) ----
        #pragma unroll
        for (int r = 0; r < 8; ++r) {
            int m = m0 + r + 8 * hi;
            lds_gi[gate * (BATCH * 16) + m * 16 + nlo] = acc_i[r] + bi;
            lds_gh[gate * (BATCH * 16) + m * 16 + nlo] = acc_h[r] + bh;
        }
        __syncthreads();

        // ---- fuse gates, update our 64x16 slice of h ----
        for (int idx = tid; idx < BATCH * 16; idx += NTHREADS) {
            int m = idx >> 4, j = idx & 15;
            float ir = lds_gi[0 * (BATCH * 16) + idx];
            float iz = lds_gi[1 * (BATCH * 16) + idx];
            float in_ = lds_gi[2 * (BATCH * 16) + idx];
            float hr = lds_gh[0 * (BATCH * 16) + idx];
            float hz = lds_gh[1 * (BATCH * 16) + idx];
            float hn = lds_gh[2 * (BATCH * 16) + idx];
            float r = 1.0f / (1.0f + __expf(-(ir + hr)));
            float z = 1.0f / (1.0f + __expf(-(iz + hz)));
            float n = tanhf(in_ + r * hn);
            float hold = lds_h[m * (HDIM + HPAD) + g * 16 + j];
            float hnew = (1.0f - z) * n + z * hold;
            hnext[(size_t)m * HDIM + g * 16 + j] = hnew;
            out[((size_t)t * BATCH + m) * HDIM + g * 16 + j] = hnew;
        }

        // ---- device-wide barrier across the 16 WGs (double-buffered h) ----
        __threadfence();          // drain this wave's stores to device scope
        __syncthreads();          // all waves arrived; LDS reuse safe
        if (tid == 0) {
            __hip_atomic_fetch_add(ctr, 1u, __ATOMIC_RELEASE,
                                   __HIP_MEMORY_SCOPE_AGENT);
            unsigned target = (unsigned)(t + 1) * NWG;
            while (__hip_atomic_load(ctr, __ATOMIC_ACQUIRE,
                                     __HIP_MEMORY_SCOPE_AGENT) < target) {
                __builtin_amdgcn_s_sleep(2);
            }
        }
        __syncthreads();
    }
}

// ---------------------------------------------------------------------------
extern "C" void kernel_launch(void* const* d_in, const int* in_sizes, int n_in,
                              void* d_out, int out_size, void* d_ws, size_t ws_size,
                              hipStream_t stream) {
    const float* x    = (const float*)d_in[0];
    const float* h0   = (const float*)d_in[1];
    const float* w_ih = (const float*)d_in[2];
    const float* w_hh = (const float*)d_in[3];
    const float* b_ih = (const float*)d_in[4];
    const float* b_hh = (const float*)d_in[5];
    float* out = (float*)d_out;

    unsigned* ctr = (unsigned*)d_ws;
    float* h0buf = (float*)((char*)d_ws + 256);
    float* h1buf = h0buf + BATCH * HDIM;

    gru_init_kernel<<<(BATCH * HDIM + 255) / 256, 256, 0, stream>>>(h0, h0buf, ctr);

    size_t smem_bytes = (size_t)(48 * (IDIM + LPAD) +        // w_ih slice
                                 48 * (HDIM + LPAD) +        // w_hh slice
                                 BATCH * (HDIM + HPAD) +     // h stage
                                 2 * 3 * BATCH * 16)         // gate exchange
                        * sizeof(float);                     // ~162 KB
    gru_scan_kernel<<<NWG, NTHREADS, smem_bytes, stream>>>(
        x, w_ih, w_hh, b_ih, b_hh, h0buf, h1buf, ctr, out);
}